// Rasterer_5669356834040
// MI455X (gfx1250) — compile-verified
//
#include <hip/hip_runtime.h>
#include <hip/hip_bf16.h>

typedef __attribute__((ext_vector_type(2))) float v2f;
typedef __attribute__((ext_vector_type(4))) float v4f;
typedef __attribute__((ext_vector_type(8))) float v8f;

#define N_VERT 1024
#define IMG_W  192
#define IMG_H  192
#define M_PIX  (IMG_W * IMG_H)      // 36864
#define CH_STR M_PIX                // channel stride in floats
#define ROW_STR (3 * M_PIX)         // vertex-row stride in floats
#define DIAM_C 0.07f
#define DEPTH_C 100.0f
#define EPS_F32 1.1920928955078125e-07f
#define N_TILES (N_VERT / 16)       // 64
#define SROW 20                     // staging row stride (floats): conflict-free
#define MSTR 17                     // mask stride (words): conflict-free

// ---------------------------------------------------------------------------
// Kernel 1: per-vertex WMMA A-operand tables, depth weights, exp table.
//   aop0[n] = {2vx, 2vy}            (A-matrix K=0,1  -> lanes 0-15)
//   aop1[n] = {r^2 - |v|^2, 1}      (A-matrix K=2,3  -> lanes 16-31)
//   w[n]    = max(0, 1 - vz/(||z||+eps)) * 100
//   e[n]    = exp(w[n] - gmax),  gmax = max_n w[n]
// ---------------------------------------------------------------------------
__global__ __launch_bounds__(N_VERT)
void build_table(const float* __restrict__ K,
                 const float* __restrict__ v2d,
                 const float* __restrict__ v3d,
                 float2* __restrict__ aop0,
                 float2* __restrict__ aop1,
                 float*  __restrict__ wtab,
                 float*  __restrict__ etab,
                 float*  __restrict__ gmax_out) {
    __shared__ float red[N_VERT];
    const int n = threadIdx.x;
    const float vz = v3d[3 * n + 2];
    red[n] = vz * vz;
    __syncthreads();
    for (int s = N_VERT / 2; s > 0; s >>= 1) {
        if (n < s) red[n] += red[n + s];
        __syncthreads();
    }
    const float znorm = sqrtf(red[0]);
    const float w = fmaxf(0.0f, 1.0f - vz / (znorm + EPS_F32)) * DEPTH_C;
    __syncthreads();                     // done reading red[0]
    red[n] = w;
    __syncthreads();
    for (int s = N_VERT / 2; s > 0; s >>= 1) {
        if (n < s) red[n] = fmaxf(red[n], red[n + s]);
        __syncthreads();
    }
    const float gmax = red[0];

    const float K00 = K[0];
    const float r  = fabsf(K00 * DIAM_C / (vz + EPS_F32));
    const float vx = v2d[2 * n + 0];
    const float vy = v2d[2 * n + 1];
    aop0[n] = make_float2(2.0f * vx, 2.0f * vy);
    aop1[n] = make_float2(r * r - vx * vx - vy * vy, 1.0f);
    wtab[n] = w;
    etab[n] = __expf(w - gmax);
    if (n == 0) gmax_out[0] = gmax;
}

// ---------------------------------------------------------------------------
// Kernel 2: one wave32 per 16 pixels.
//  Pass A: 64 x V_WMMA_F32_16X16X4_F32 -> D[n,m] = r^2 - dist^2 ; branchless
//          bit-pack, one word per 4 tiles, words parked in LDS (no spills).
//  Pass B: sparse ffs walk over set bits -> masked max(w) and sum of e.
//  Pass C: per tile, stage the 16x16 value block in LDS (row stride 20),
//          then 2 x ds_load_b128 per lane + 3 NT global_store_b128 each.
// ---------------------------------------------------------------------------
__global__ __launch_bounds__(256, 4)
void raster_main(const float2* __restrict__ aop0,
                 const float2* __restrict__ aop1,
                 const float*  __restrict__ wtab,
                 const float*  __restrict__ etab,
                 const float*  __restrict__ gmax_in,
                 float* __restrict__ out) {
    __shared__ __align__(16) float2 sA[2][N_VERT];     // 16 KB
    __shared__ float sw[N_VERT];                       //  4 KB
    __shared__ __align__(16) float se[N_VERT];         //  4 KB
    __shared__ __align__(16) float sval[8][16 * SROW]; // 10 KB staging (per wave)
    __shared__ unsigned smask[256 * MSTR];             // 17 KB per-thread masks

    for (int i = threadIdx.x; i < N_VERT; i += 256) {
        sA[0][i] = aop0[i];
        sA[1][i] = aop1[i];
    }
    {
        const int i = threadIdx.x;                   // 256 float4 = 1024 floats
        ((float4*)sw)[i] = ((const float4*)wtab)[i];
        ((float4*)se)[i] = ((const float4*)etab)[i];
    }
    const float gmax = gmax_in[0];                   // uniform -> s_load
    __syncthreads();

    const int lane = threadIdx.x & 31;
    const int hi   = lane >> 4;          // 0: K={0,1}, rows v ; 1: K={2,3}, rows v+8
    const int col  = lane & 15;          // pixel column within tile
    const int wave = threadIdx.x >> 5;
    const int p0   = (blockIdx.x * 8 + wave) * 16;   // wave's base pixel
    const int p    = p0 + col;
    const float px = (float)(p % IMG_W);
    const float py = (float)(p / IMG_W);

    // B matrix (4 x 16): rows {px, py, 1, -|p|^2}
    v2f bm;
    bm.x = hi ? 1.0f : px;
    bm.y = hi ? -(px * px + py * py) : py;

    v8f cz = {};
    const float2* sAh = sA[hi];
    unsigned* const mbase = &smask[threadIdx.x * MSTR];

    // -------- Pass A: WMMA distance tiles, bit-pack one word per 4 tiles ----
#pragma unroll
    for (int i = 0; i < 16; ++i) {
        unsigned mw = 0u;
#pragma unroll
        for (int tt = 0; tt < 4; ++tt) {
            const int t = i * 4 + tt;
            const float2 a2 = sAh[t * 16 + col];     // one ds_load_b64
            v2f am;
            am.x = a2.x;
            am.y = a2.y;
            v8f d = __builtin_amdgcn_wmma_f32_16x16x4_f32(
                false, am, false, bm, (short)0, cz, false, false);
#pragma unroll
            for (int v = 0; v < 8; ++v) {
                mw |= (d[v] > 0.0f) ? (1u << ((tt << 3) | v)) : 0u;
            }
            __builtin_amdgcn_sched_barrier(0);       // cap live ranges / VGPRs
        }
        mbase[i] = mw;                               // park word in LDS
    }

    // -------- Pass B: sparse stats over set bits --------
    float maxw = 0.0f;       // unmasked entries contribute value 0 to softmax
    float sume = 0.0f;
    int   cnt  = 0;
#pragma unroll
    for (int i = 0; i < 16; ++i) {
        unsigned wrd = mbase[i];
        cnt += __popc(wrd);
        while (wrd) {
            const int b = __ffs(wrd) - 1;
            wrd &= wrd - 1;
            const int t = i * 4 + (b >> 3);
            const int v = b & 7;
            const int row = t * 16 + v + 8 * hi;
            maxw = fmaxf(maxw, sw[row]);
            sume += se[row];
        }
    }
    // combine the two lane-halves sharing this pixel column
    const float maxp = fmaxf(maxw, __shfl_xor(maxw, 16, 32));
    const float sumE = sume + __shfl_xor(sume, 16, 32);
    const int   cntp = cnt  + __shfl_xor(cnt, 16, 32);

    const float eg = __expf(gmax - maxp);            // finite whenever any bit set
    const float S  = eg * sumE + (float)(N_VERT - cntp) * __expf(-maxp);
    const float Cs = eg / S;                         // exp(w-maxp) = e * eg

    // -------- Pass C: stage 16x16 block in LDS, write b128 NT stores --------
    const float4* se4 = (const float4*)se;
    float* const sv_st = &sval[wave][(8 * hi) * SROW + col];   // store base
    const int cid0 = lane;               // chunk ids lane, lane+32
#pragma unroll
    for (int i = 0; i < 16; ++i) {
        const unsigned mw = mbase[i];                // one ds_load per 4 tiles
#pragma unroll
        for (int tt = 0; tt < 4; ++tt) {
            const int t = i * 4 + tt;
            const float4 e0 = se4[t * 4 + 2 * hi];       // e for rows base+0..3
            const float4 e1 = se4[t * 4 + 2 * hi + 1];   // e for rows base+4..7
            float es[8];
            es[0] = e0.x * Cs; es[1] = e0.y * Cs; es[2] = e0.z * Cs; es[3] = e0.w * Cs;
            es[4] = e1.x * Cs; es[5] = e1.y * Cs; es[6] = e1.z * Cs; es[7] = e1.w * Cs;
            // stage this lane's 8 values (rows 8*hi+v, column col)
#pragma unroll
            for (int v = 0; v < 8; ++v) {
                const float val = ((mw >> ((tt << 3) | v)) & 1u) ? es[v] : 0.0f;
                sv_st[v * SROW] = val;
            }
            // gather 4 consecutive pixels of one row per lane, 2 chunks per lane
#pragma unroll
            for (int j = 0; j < 2; ++j) {
                const int cid = cid0 + j * 32;           // 0..63
                const int r   = cid >> 2;                // row within tile 0..15
                const int q   = (cid & 3) << 2;          // pixel sub-offset 0/4/8/12
                const v4f vv = *(const v4f*)&sval[wave][r * SROW + q];
                float* gp = out + (size_t)(t * 16 + r) * ROW_STR + (p0 + q);
                __builtin_nontemporal_store(vv, (v4f*)gp);
                __builtin_nontemporal_store(vv, (v4f*)(gp + CH_STR));
                __builtin_nontemporal_store(vv, (v4f*)(gp + 2 * CH_STR));
            }
            __builtin_amdgcn_sched_barrier(0);       // cap live ranges / VGPRs
        }
    }
}

// ---------------------------------------------------------------------------
extern "C" void kernel_launch(void* const* d_in, const int* in_sizes, int n_in,
                              void* d_out, int out_size, void* d_ws, size_t ws_size,
                              hipStream_t stream) {
    const float* K   = (const float*)d_in[0];
    // d_in[1] = grid_2d (regenerated from pixel index), d_in[4] = normals (unused)
    const float* v2d = (const float*)d_in[2];
    const float* v3d = (const float*)d_in[3];

    float2* aop0 = (float2*)d_ws;                          //  8 KB
    float2* aop1 = (float2*)((char*)d_ws + 8192);          //  8 KB
    float*  wtab = (float*)((char*)d_ws + 16384);          //  4 KB
    float*  etab = (float*)((char*)d_ws + 20480);          //  4 KB
    float*  gmaxp = (float*)((char*)d_ws + 24576);         //  4 B

    build_table<<<1, N_VERT, 0, stream>>>(K, v2d, v3d, aop0, aop1, wtab, etab, gmaxp);

    const int groups = M_PIX / 16;          // 2304 waves
    const int blocks = groups / 8;          // 288 blocks x 256 threads
    raster_main<<<blocks, 256, 0, stream>>>(aop0, aop1, wtab, etab, gmaxp, (float*)d_out);
}